// GNN_58025008168970
// MI455X (gfx1250) — compile-verified
//
#include <hip/hip_runtime.h>
#include <hip/hip_bf16.h>

#define N_NODES   50000
#define N_EDGES   600000
#define N_GRAPHS  256
#define DF        128
#define D_OUT     32
#define NEG_SLOPE 0.01f
#define EPS_NORM  1e-12f

typedef __attribute__((ext_vector_type(2))) float v2f;
typedef __attribute__((ext_vector_type(8))) float v8f;

// ---------------------------------------------------------------------------
// Utility fills
// ---------------------------------------------------------------------------
__global__ void fill_kernel(float* __restrict__ p, float v, int n) {
    int i = blockIdx.x * blockDim.x + threadIdx.x;
    if (i < n) p[i] = v;
}

// deg[dst] += 1 per edge (deg pre-filled with 1.0 for the self loop)
__global__ void deg_count_kernel(const int* __restrict__ dst, float* __restrict__ deg) {
    int e = blockIdx.x * blockDim.x + threadIdx.x;
    if (e < N_EDGES) atomicAdd(&deg[dst[e]], 1.0f);
}

__global__ void deg_finalize_kernel(const float* __restrict__ deg,
                                    float* __restrict__ isd,
                                    float* __restrict__ invd) {
    int i = blockIdx.x * blockDim.x + threadIdx.x;
    if (i < N_NODES) {
        float d = deg[i];
        isd[i]  = rsqrtf(d);
        invd[i] = 1.0f / d;
    }
}

// ---------------------------------------------------------------------------
// C[N,128] = A[N,128] @ W[128,128] (+ bias)   via V_WMMA_F32_16X16X4_F32
//
// Block: 128 threads (4 waves). Wave w handles rows [(4*bx+w)*16, +16),
// producing all 8 column tiles of 16.
//
// W is staged TRANSPOSED in LDS: ldsT[col * 132 + k]. Pitch 132 (= 4 mod 64
// dwords) makes the per-lane ds_load_b64 of (k, k+1) hit all 64 banks exactly
// once across the wave (conflict-free), and gives the B fragment as a single
// 64-bit LDS load into an even-aligned VGPR pair.
//
// Fragment layouts (ISA 7.12.2, 32-bit, wave32):
//   A 16x4 : lane -> M = lane%16, K = k0 + 2*(lane/16) + v
//   B 4x16 : lane -> N = lane%16, K = k0 + 2*(lane/16) + v
//   C/D    : vgpr r -> M = r + 8*(lane/16), N = lane%16
// ---------------------------------------------------------------------------
#define LDS_T_PITCH 132   // dwords per LDS column-row; 128*132*4 = 67584 B

__global__ void __launch_bounds__(128)
gemm128_wmma_kernel(const float* __restrict__ A, const float* __restrict__ W,
                    const float* __restrict__ bias, float* __restrict__ C,
                    int n_rows) {
    extern __shared__ float ldsT[];
    const int tid  = threadIdx.x;
    const int wave = tid >> 5;
    const int lane = tid & 31;

    // Stage W transposed: thread i handles W[r][c4..c4+3] with r = i&31 so
    // that LDS store banks (4c + r) mod 64 are distinct within a wave.
    for (int i = tid; i < 128 * 32; i += 128) {
        const int r  = i & 31;
        const int c4 = (i >> 5) << 2;
        const float4 v = *(const float4*)(W + (size_t)r * DF + c4);
        ldsT[(c4 + 0) * LDS_T_PITCH + r] = v.x;
        ldsT[(c4 + 1) * LDS_T_PITCH + r] = v.y;
        ldsT[(c4 + 2) * LDS_T_PITCH + r] = v.z;
        ldsT[(c4 + 3) * LDS_T_PITCH + r] = v.w;
    }
    __syncthreads();

    const int rowBase = (blockIdx.x * 4 + wave) * 16;
    if (rowBase >= n_rows) return;          // uniform per wave: EXEC stays full

    const int m  = lane & 15;               // row (A) / col (B) within tile
    const int hi = lane >> 4;               // half-wave selector

    v8f acc[8];
    {
        v8f z = {0.f, 0.f, 0.f, 0.f, 0.f, 0.f, 0.f, 0.f};
        #pragma unroll
        for (int t = 0; t < 8; ++t) acc[t] = z;
    }

    const float* arow = A + (size_t)(rowBase + m) * DF;
    for (int k0 = 0; k0 < DF; k0 += 4) {
        const int ka = k0 + (hi << 1);                 // even -> 8B aligned
        const v2f a = *(const v2f*)(arow + ka);        // global_load_b64
        #pragma unroll
        for (int t = 0; t < 8; ++t) {
            const int col = (t << 4) + m;
            const v2f b = *(const v2f*)(&ldsT[col * LDS_T_PITCH + ka]); // ds_load_b64
            acc[t] = __builtin_amdgcn_wmma_f32_16x16x4_f32(
                false, a, false, b, (short)0, acc[t], false, false);
        }
    }

    #pragma unroll
    for (int t = 0; t < 8; ++t) {
        const int col  = (t << 4) + m;
        const float bv = bias ? bias[col] : 0.0f;
        #pragma unroll
        for (int r = 0; r < 8; ++r) {
            const int row = rowBase + r + (hi << 3);
            C[(size_t)row * DF + col] = acc[t][r] + bv;
        }
    }
}

// ---------------------------------------------------------------------------
// agg[i,:] = ht[i,:] * inv_deg[i] + b[:]   (self-loop term seeds the buffer)
// ---------------------------------------------------------------------------
__global__ void init_agg_kernel(const float* __restrict__ ht,
                                const float* __restrict__ invd,
                                const float* __restrict__ b,
                                float* __restrict__ agg) {
    int t = blockIdx.x * blockDim.x + threadIdx.x;     // over N_NODES * 32
    int i = t >> 5;
    if (i >= N_NODES) return;
    int lane = t & 31;
    float id4 = invd[i];
    float4 v = ((const float4*)(ht + (size_t)i * DF))[lane];
    const float4 bb = ((const float4*)b)[lane];
    float4 o;
    o.x = v.x * id4 + bb.x;
    o.y = v.y * id4 + bb.y;
    o.z = v.z * id4 + bb.z;
    o.w = v.w * id4 + bb.w;
    ((float4*)(agg + (size_t)i * DF))[lane] = o;
}

// ---------------------------------------------------------------------------
// agg[dst,:] += isd[src]*isd[dst] * ht[src,:]   (one edge per 32 lanes)
// ---------------------------------------------------------------------------
__global__ void scatter_kernel(const float* __restrict__ ht,
                               const int* __restrict__ src,
                               const int* __restrict__ dst,
                               const float* __restrict__ isd,
                               float* __restrict__ agg) {
    int t = blockIdx.x * blockDim.x + threadIdx.x;     // over N_EDGES * 32
    int e = t >> 5;
    if (e >= N_EDGES) return;
    int lane = t & 31;
    int s = src[e], d = dst[e];
    float coef = isd[s] * isd[d];
    float4 v = ((const float4*)(ht + (size_t)s * DF))[lane];
    float* ap = agg + (size_t)d * DF + (lane << 2);
    atomicAdd(ap + 0, v.x * coef);
    atomicAdd(ap + 1, v.y * coef);
    atomicAdd(ap + 2, v.z * coef);
    atomicAdd(ap + 3, v.w * coef);
}

__global__ void leaky_kernel(float* __restrict__ p, int n) {
    int i = blockIdx.x * blockDim.x + threadIdx.x;
    if (i < n) {
        float v = p[i];
        p[i] = v > 0.0f ? v : v * NEG_SLOPE;
    }
}

// ---------------------------------------------------------------------------
// sums[batch[i],:] += h[i,:];  counts[batch[i]] += 1
// ---------------------------------------------------------------------------
__global__ void pool_kernel(const float* __restrict__ h,
                            const int* __restrict__ batch,
                            float* __restrict__ sums,
                            float* __restrict__ counts) {
    int t = blockIdx.x * blockDim.x + threadIdx.x;     // over N_NODES * 32
    int i = t >> 5;
    if (i >= N_NODES) return;
    int lane = t & 31;
    int g = batch[i];
    float4 v = ((const float4*)(h + (size_t)i * DF))[lane];
    float* sp = sums + (size_t)g * DF + (lane << 2);
    atomicAdd(sp + 0, v.x);
    atomicAdd(sp + 1, v.y);
    atomicAdd(sp + 2, v.z);
    atomicAdd(sp + 3, v.w);
    if (lane == 0) atomicAdd(&counts[g], 1.0f);
}

// ---------------------------------------------------------------------------
// Per-graph head: mean -> L2 normalize -> leaky(fc2) -> fc3. 1 block / graph.
// ---------------------------------------------------------------------------
__global__ void __launch_bounds__(128)
head_kernel(const float* __restrict__ sums, const float* __restrict__ counts,
            const float* __restrict__ W2, const float* __restrict__ b2,
            const float* __restrict__ W3, const float* __restrict__ b3,
            float* __restrict__ out) {
    __shared__ float hg[DF];
    __shared__ float red[DF];
    __shared__ float h2[DF];
    const int g = blockIdx.x;
    const int t = threadIdx.x;

    float cnt = counts[g];
    cnt = cnt > 1.0f ? cnt : 1.0f;
    float p = sums[(size_t)g * DF + t] / cnt;

    red[t] = p * p;
    __syncthreads();
    for (int s = 64; s > 0; s >>= 1) {
        if (t < s) red[t] += red[t + s];
        __syncthreads();
    }
    float nrm = sqrtf(red[0]);
    nrm = nrm > EPS_NORM ? nrm : EPS_NORM;
    hg[t] = p / nrm;
    __syncthreads();

    float acc = b2[t];
    for (int k = 0; k < DF; ++k) acc += hg[k] * W2[k * DF + t];
    acc = acc > 0.0f ? acc : acc * NEG_SLOPE;
    h2[t] = acc;
    __syncthreads();

    if (t < D_OUT) {
        float o = b3[t];
        for (int k = 0; k < DF; ++k) o += h2[k] * W3[k * D_OUT + t];
        out[(size_t)g * D_OUT + t] = o;
    }
}

// ---------------------------------------------------------------------------
extern "C" void kernel_launch(void* const* d_in, const int* in_sizes, int n_in,
                              void* d_out, int out_size, void* d_ws, size_t ws_size,
                              hipStream_t stream) {
    const float* x      = (const float*)d_in[0];
    const int*   ei     = (const int*)  d_in[1];   // [2, E]
    const int*   batch  = (const int*)  d_in[2];
    const float* W_fc1  = (const float*)d_in[3];
    const float* b_fc1  = (const float*)d_in[4];
    const float* W_gc1  = (const float*)d_in[5];
    const float* b_gc1  = (const float*)d_in[6];
    const float* W_gc2  = (const float*)d_in[7];
    const float* b_gc2  = (const float*)d_in[8];
    const float* W_fc2  = (const float*)d_in[9];
    const float* b_fc2  = (const float*)d_in[10];
    const float* W_fc3  = (const float*)d_in[11];
    const float* b_fc3  = (const float*)d_in[12];
    float* out = (float*)d_out;

    const int* src = ei;
    const int* dst = ei + N_EDGES;

    // Workspace layout
    float* ws     = (float*)d_ws;
    float* deg    = ws;                              // N
    float* isd    = deg  + N_NODES;                  // N
    float* invd   = isd  + N_NODES;                  // N
    float* bufA   = invd + N_NODES;                  // N*DF
    float* bufB   = bufA + (size_t)N_NODES * DF;     // N*DF
    float* bufC   = bufB + (size_t)N_NODES * DF;     // N*DF
    float* sums   = bufC + (size_t)N_NODES * DF;     // G*DF
    float* counts = sums + (size_t)N_GRAPHS * DF;    // G

    const int TB = 256;
    const size_t gemm_lds = (size_t)128 * LDS_T_PITCH * sizeof(float);  // 67584 B
    const int gemm_blocks = (N_NODES / 16 + 3) / 4;                     // 782

    // --- degrees ---
    fill_kernel<<<(N_NODES + TB - 1) / TB, TB, 0, stream>>>(deg, 1.0f, N_NODES);
    deg_count_kernel<<<(N_EDGES + TB - 1) / TB, TB, 0, stream>>>(dst, deg);
    deg_finalize_kernel<<<(N_NODES + TB - 1) / TB, TB, 0, stream>>>(deg, isd, invd);

    // --- fc1: bufA = x @ W_fc1 + b_fc1 ---
    gemm128_wmma_kernel<<<gemm_blocks, 128, gemm_lds, stream>>>(x, W_fc1, b_fc1, bufA, N_NODES);

    // --- GCN layer 1 ---
    gemm128_wmma_kernel<<<gemm_blocks, 128, gemm_lds, stream>>>(bufA, W_gc1, nullptr, bufB, N_NODES);
    init_agg_kernel<<<(N_NODES * 32 + TB - 1) / TB, TB, 0, stream>>>(bufB, invd, b_gc1, bufC);
    scatter_kernel<<<(N_EDGES * 32 + TB - 1) / TB, TB, 0, stream>>>(bufB, src, dst, isd, bufC);
    leaky_kernel<<<((size_t)N_NODES * DF + TB - 1) / TB, TB, 0, stream>>>(bufC, N_NODES * DF);

    // --- GCN layer 2 ---
    gemm128_wmma_kernel<<<gemm_blocks, 128, gemm_lds, stream>>>(bufC, W_gc2, nullptr, bufA, N_NODES);
    init_agg_kernel<<<(N_NODES * 32 + TB - 1) / TB, TB, 0, stream>>>(bufA, invd, b_gc2, bufB);
    scatter_kernel<<<(N_EDGES * 32 + TB - 1) / TB, TB, 0, stream>>>(bufA, src, dst, isd, bufB);
    leaky_kernel<<<((size_t)N_NODES * DF + TB - 1) / TB, TB, 0, stream>>>(bufB, N_NODES * DF);

    // --- mean pool ---
    fill_kernel<<<(N_GRAPHS * DF + N_GRAPHS + TB - 1) / TB, TB, 0, stream>>>(sums, 0.0f,
                                                                             N_GRAPHS * DF + N_GRAPHS);
    pool_kernel<<<(N_NODES * 32 + TB - 1) / TB, TB, 0, stream>>>(bufB, batch, sums, counts);

    // --- normalize + fc2 + fc3 ---
    head_kernel<<<N_GRAPHS, 128, 0, stream>>>(sums, counts, W_fc2, b_fc2, W_fc3, b_fc3, out);
}